// PPOLoss_13898514170707
// MI455X (gfx1250) — compile-verified
//
#include <hip/hip_runtime.h>
#include <math.h>

// ---------------- constants ----------------
#define TPB   256          // threads per block (8 waves of 32)
#define EPT   32           // elements per thread (serial reverse-Horner)
#define CHUNK (TPB * EPT)  // 8192 elements per block
#define MAXB  1024         // max blocks (T = 2^23 -> exactly 1024)

#define A1C 0.9405f        // gamma * lambda
#define A2C 0.99f          // gamma
#define GAM 0.99f
#define EPSC 0.2f

typedef __attribute__((ext_vector_type(2))) float v2f;
typedef __attribute__((ext_vector_type(8))) float v8f;

// ---------------------------------------------------------------------------
// 16x16 constant-ratio scan of 16 segments via WMMA f32 16x16x4 chain.
// Mh[256] in LDS holds the triangular power matrix M[m][k] = A32^(k-m), k>=m.
// sv[256] in LDS holds the data; segment n = sv[16n .. 16n+15].
// Output D[i][n] = sum_{j>=i} A32^(j-i) * sv[16n + j]  (zero-carry local scan)
// as a 16x16 f32 C/D matrix (8 VGPRs/lane, wave32 layout per ISA 7.12.2).
// Must be called by a full wave (EXEC all ones).
// ---------------------------------------------------------------------------
__device__ __forceinline__ v8f scan16x16_wmma(const float* sv, const float* Mh) {
  const int L  = threadIdx.x & 31;
  const int hi = L >> 4;   // half-wave: selects K pair {0,1} vs {2,3}
  const int lo = L & 15;   // A: row m ; B: column n
  v8f c = {0.f, 0.f, 0.f, 0.f, 0.f, 0.f, 0.f, 0.f};
#pragma unroll
  for (int p = 0; p < 4; ++p) {           // K-slices of the 16x16 matmul
    const int k0 = 4 * p + 2 * hi;
    const int k1 = k0 + 1;
    v2f a, b;
    a.x = Mh[lo * 16 + k0];               // A[m=lo][k0]
    a.y = Mh[lo * 16 + k1];
    b.x = sv[lo * 16 + k0];               // B[k][n=lo] = s[16n + k]
    b.y = sv[lo * 16 + k1];
    c = __builtin_amdgcn_wmma_f32_16x16x4_f32(
        /*neg_a=*/false, a, /*neg_b=*/false, b,
        /*c_mod=*/(short)0, c, /*reuse_a=*/false, /*reuse_b=*/false);
  }
  return c;
}

// ---------------------------------------------------------------------------
// Kernel A: per-thread weighted partials (reverse Horner over 32 elems) and
// per-block weighted partials P_b = sum_t A32^t * s_t for both recurrences.
// ---------------------------------------------------------------------------
__global__ __launch_bounds__(TPB) void ppo_partials(
    const float* __restrict__ rw, const float* __restrict__ vl,
    float* __restrict__ sT1, float* __restrict__ sT2,
    float* __restrict__ P1, float* __restrict__ P2,
    float l2A32_1, float l2A32_2, int T) {
  const int t = threadIdx.x, b = blockIdx.x;
  const int base = b * CHUNK + t * EPT;
  float y1 = 0.f, y2 = 0.f;
  if (base < T) {
    const float4* r4p = reinterpret_cast<const float4*>(rw + base);
    const float4* v4p = reinterpret_cast<const float4*>(vl + base);
    float prevVx = (base + EPT < T) ? vl[base + EPT] : 0.f;  // v[seg_end]
#pragma unroll
    for (int q = 7; q >= 0; --q) {   // j = 31 .. 0 in quads
      const float4 r4 = r4p[q];
      const float4 v4 = v4p[q];
      y1 = fmaf(A1C, y1, r4.w - v4.w + GAM * prevVx); y2 = fmaf(A2C, y2, r4.w);
      y1 = fmaf(A1C, y1, r4.z - v4.z + GAM * v4.w);   y2 = fmaf(A2C, y2, r4.z);
      y1 = fmaf(A1C, y1, r4.y - v4.y + GAM * v4.z);   y2 = fmaf(A2C, y2, r4.y);
      y1 = fmaf(A1C, y1, r4.x - v4.x + GAM * v4.y);   y2 = fmaf(A2C, y2, r4.x);
      prevVx = v4.x;
    }
  }
  const int gt = b * TPB + t;
  sT1[gt] = y1;
  sT2[gt] = y2;

  __shared__ float red1[TPB], red2[TPB];
  red1[t] = exp2f((float)t * l2A32_1) * y1;   // weight A32^t (underflows -> 0)
  red2[t] = exp2f((float)t * l2A32_2) * y2;
  __syncthreads();
  for (int s = TPB / 2; s > 0; s >>= 1) {
    if (t < s) { red1[t] += red1[t + s]; red2[t] += red2[t + s]; }
    __syncthreads();
  }
  if (t == 0) { P1[b] = red1[0]; P2[b] = red2[0]; }
}

// ---------------------------------------------------------------------------
// Kernel B: inter-block carry scan.  y_b = sum_{u>=b} AC^(u-b) P_u via
// constant-ratio Hillis-Steele; carryIn[b] = y_{b+1}.
// ---------------------------------------------------------------------------
__global__ __launch_bounds__(MAXB) void ppo_carry(
    const float* __restrict__ P1, const float* __restrict__ P2,
    float* __restrict__ C1, float* __restrict__ C2,
    int B, float AC1, float AC2) {
  __shared__ float y1[MAXB], y2[MAXB];
  const int t = threadIdx.x;
  y1[t] = (t < B) ? P1[t] : 0.f;
  y2[t] = (t < B) ? P2[t] : 0.f;
  __syncthreads();
  float W1 = AC1, W2 = AC2;            // AC^len, squared each doubling
  for (int off = 1; off < MAXB; off <<= 1) {
    const float n1 = (t + off < MAXB) ? y1[t + off] : 0.f;
    const float n2 = (t + off < MAXB) ? y2[t + off] : 0.f;
    __syncthreads();
    y1[t] += W1 * n1;
    y2[t] += W2 * n2;
    __syncthreads();
    W1 *= W1; W2 *= W2;
  }
  if (t < B) {
    C1[t] = (t + 1 < B) ? y1[t + 1] : 0.f;
    C2[t] = (t + 1 < B) ? y2[t + 1] : 0.f;
  }
}

// ---------------------------------------------------------------------------
// Kernel C: main fused pass.  Wave 0 runs the WMMA segment scans over the 256
// thread partials, segment heads are chained serially, every thread gets its
// exact carry V_{t+1}, then streams its 32 elements in reverse computing
// advantages, targets, and all three loss terms.  Deterministic block sum.
// ---------------------------------------------------------------------------
__global__ __launch_bounds__(TPB) void ppo_main(
    const float* __restrict__ pr, const float* __restrict__ po,
    const float* __restrict__ rw, const float* __restrict__ vl,
    const float* __restrict__ sT1, const float* __restrict__ sT2,
    const float* __restrict__ C1, const float* __restrict__ C2,
    float* __restrict__ blockSums,
    float l2A32_1, float l2A32_2, float A512_1, float A512_2, int T) {
  __shared__ float s1sh[TPB], s2sh[TPB];
  __shared__ float M1sh[TPB], M2sh[TPB];   // triangular power matrices
  __shared__ float Y1sh[TPB], Y2sh[TPB];
  __shared__ float G1sh[17], G2sh[17];
  __shared__ float red[TPB];

  const int t = threadIdx.x, b = blockIdx.x;
  const int gt = b * TPB + t;
  s1sh[t] = sT1[gt];
  s2sh[t] = sT2[gt];
  // Stage M[m][k] = A32^(k-m) for k>=m else 0 (one exp2f per thread, no loops)
  {
    const int m = t >> 4, k = t & 15;
    const float e = (float)(k - m);
    M1sh[t] = (k >= m) ? exp2f(e * l2A32_1) : 0.f;
    M2sh[t] = (k >= m) ? exp2f(e * l2A32_2) : 0.f;
  }
  __syncthreads();

  // --- level 2: 16-wide segment scans on the matrix pipe (wave 0 only) ---
  if (t < 32) {
    const v8f c1 = scan16x16_wmma(s1sh, M1sh);
    const v8f c2 = scan16x16_wmma(s2sh, M2sh);
    const int hi = t >> 4, lo = t & 15;
#pragma unroll
    for (int vi = 0; vi < 8; ++vi) {     // C/D layout: VGPR vi = rows vi, vi+8
      Y1sh[(vi + 8 * hi) * 16 + lo] = c1[vi];
      Y2sh[(vi + 8 * hi) * 16 + lo] = c2[vi];
    }
  }
  __syncthreads();

  // --- combine 16 segment heads with the chunk carry-in (16 serial steps) ---
  if (t == 0) {
    float g1 = C1[b], g2 = C2[b];
    G1sh[16] = g1; G2sh[16] = g2;
    for (int m = 15; m >= 0; --m) {
      g1 = Y1sh[m] + A512_1 * g1; G1sh[m] = g1;   // Y[0][m]
      g2 = Y2sh[m] + A512_2 * g2; G2sh[m] = g2;
    }
  }
  __syncthreads();

  // --- each thread derives its exact carry V_{t+1} (branch-free form):
  //     V_{t+1} = (i<15 ? Y[i+1][n] : 0) + A32^(15-i) * G[n+1]   (A32^0 = 1)
  const int n = t >> 4, i = t & 15;
  const float wtail1 = exp2f((float)(15 - i) * l2A32_1);
  const float wtail2 = exp2f((float)(15 - i) * l2A32_2);
  const float sel1 = (i < 15) ? Y1sh[(i + 1) * 16 + n] : 0.f;
  const float sel2 = (i < 15) ? Y2sh[(i + 1) * 16 + n] : 0.f;
  float y1 = sel1 + wtail1 * G1sh[n + 1];
  float y2 = sel2 + wtail2 * G2sh[n + 1];

  // --- level 1: final reverse stream, fused loss evaluation ---
  float acc = 0.f;
  const int base = b * CHUNK + t * EPT;
  if (base < T) {
    const float4* p4p  = reinterpret_cast<const float4*>(pr + base);
    const float4* po4p = reinterpret_cast<const float4*>(po + base);
    const float4* r4p  = reinterpret_cast<const float4*>(rw + base);
    const float4* v4p  = reinterpret_cast<const float4*>(vl + base);
    float prevVx = (base + EPT < T) ? vl[base + EPT] : 0.f;
    const float lo_clip = 1.f - EPSC, hi_clip = 1.f + EPSC;
#pragma unroll
    for (int q = 7; q >= 0; --q) {
      const float4 p4 = p4p[q], q4 = po4p[q], r4 = r4p[q], v4 = v4p[q];
      // j = 4q+3 .. 4q
      {
        y1 = fmaf(A1C, y1, r4.w - v4.w + GAM * prevVx);
        y2 = fmaf(A2C, y2, r4.w);
        const float ra = p4.w / q4.w;
        const float cl = fminf(fmaxf(ra, lo_clip), hi_clip);
        const float dv = v4.w - y2;
        acc += -fminf(ra * y1, cl * y1) + 0.5f * dv * dv
               - 0.01f * (p4.w * __logf(p4.w + 1e-5f));
      }
      {
        y1 = fmaf(A1C, y1, r4.z - v4.z + GAM * v4.w);
        y2 = fmaf(A2C, y2, r4.z);
        const float ra = p4.z / q4.z;
        const float cl = fminf(fmaxf(ra, lo_clip), hi_clip);
        const float dv = v4.z - y2;
        acc += -fminf(ra * y1, cl * y1) + 0.5f * dv * dv
               - 0.01f * (p4.z * __logf(p4.z + 1e-5f));
      }
      {
        y1 = fmaf(A1C, y1, r4.y - v4.y + GAM * v4.z);
        y2 = fmaf(A2C, y2, r4.y);
        const float ra = p4.y / q4.y;
        const float cl = fminf(fmaxf(ra, lo_clip), hi_clip);
        const float dv = v4.y - y2;
        acc += -fminf(ra * y1, cl * y1) + 0.5f * dv * dv
               - 0.01f * (p4.y * __logf(p4.y + 1e-5f));
      }
      {
        y1 = fmaf(A1C, y1, r4.x - v4.x + GAM * v4.y);
        y2 = fmaf(A2C, y2, r4.x);
        const float ra = p4.x / q4.x;
        const float cl = fminf(fmaxf(ra, lo_clip), hi_clip);
        const float dv = v4.x - y2;
        acc += -fminf(ra * y1, cl * y1) + 0.5f * dv * dv
               - 0.01f * (p4.x * __logf(p4.x + 1e-5f));
      }
      prevVx = v4.x;
    }
  }

  red[t] = acc;
  __syncthreads();
  for (int s = TPB / 2; s > 0; s >>= 1) {
    if (t < s) red[t] += red[t + s];
    __syncthreads();
  }
  if (t == 0) blockSums[b] = red[0];
}

// ---------------------------------------------------------------------------
// Kernel D: deterministic final reduction of per-block sums.
// ---------------------------------------------------------------------------
__global__ __launch_bounds__(MAXB) void ppo_final(
    const float* __restrict__ bs, float* __restrict__ out, int B) {
  __shared__ float red[MAXB];
  const int t = threadIdx.x;
  red[t] = (t < B) ? bs[t] : 0.f;
  __syncthreads();
  for (int s = MAXB / 2; s > 0; s >>= 1) {
    if (t < s) red[t] += red[t + s];
    __syncthreads();
  }
  if (t == 0) out[0] = red[0];
}

// ---------------------------------------------------------------------------
extern "C" void kernel_launch(void* const* d_in, const int* in_sizes, int n_in,
                              void* d_out, int out_size, void* d_ws, size_t ws_size,
                              hipStream_t stream) {
  const float* probs     = (const float*)d_in[0];
  const float* probs_old = (const float*)d_in[1];
  const float* rewards   = (const float*)d_in[2];
  const float* values    = (const float*)d_in[3];
  const int T = in_sizes[0];
  int B = (T + CHUNK - 1) / CHUNK;
  if (B > MAXB) B = MAXB;  // T = 2^23 -> exactly 1024

  float* ws  = (float*)d_ws;
  float* sT1 = ws;                          // B*TPB thread partials (GAE)
  float* sT2 = sT1 + (size_t)B * TPB;       // B*TPB thread partials (targets)
  float* P1  = sT2 + (size_t)B * TPB;       // B block partials
  float* P2  = P1 + B;
  float* C1  = P2 + B;                      // B carries
  float* C2  = C1 + B;
  float* BS  = C2 + B;                      // B block loss sums

  const double a1 = 0.99 * 0.95, a2 = 0.99;
  const float AC1 = (float)pow(a1, (double)CHUNK);     // -> 0.0f (underflow)
  const float AC2 = (float)pow(a2, (double)CHUNK);     // ~2e-36
  const float l2A32_1 = (float)(32.0 * log2(a1));      // log2(a1^32)
  const float l2A32_2 = (float)(32.0 * log2(a2));      // log2(a2^32)
  const float A512_1  = (float)pow(a1, 512.0);         // (a1^32)^16
  const float A512_2  = (float)pow(a2, 512.0);         // (a2^32)^16

  ppo_partials<<<B, TPB, 0, stream>>>(rewards, values, sT1, sT2, P1, P2,
                                      l2A32_1, l2A32_2, T);
  ppo_carry<<<1, MAXB, 0, stream>>>(P1, P2, C1, C2, B, AC1, AC2);
  ppo_main<<<B, TPB, 0, stream>>>(probs, probs_old, rewards, values,
                                  sT1, sT2, C1, C2, BS,
                                  l2A32_1, l2A32_2, A512_1, A512_2, T);
  ppo_final<<<1, MAXB, 0, stream>>>(BS, (float*)d_out, B);
}